// GCN_23845658428197
// MI455X (gfx1250) — compile-verified
//
#include <hip/hip_runtime.h>

typedef __attribute__((ext_vector_type(2))) float v2f;
typedef __attribute__((ext_vector_type(8))) float v8f;

#define DCH 128   // channels (in = hidden = out)

// ---------------- degree / normalization ----------------

__global__ void gcn_deg_init(float* deg, int n) {
    int i = blockIdx.x * blockDim.x + threadIdx.x;
    if (i < n) deg[i] = 1.0f;              // self-loop contributes 1
}

__global__ void gcn_deg_count(const int* __restrict__ tgt, float* deg, int e) {
    int i = blockIdx.x * blockDim.x + threadIdx.x;
    if (i < e) atomicAdd(&deg[tgt[i]], 1.0f);
}

__global__ void gcn_rsqrt(float* deg, int n) {
    int i = blockIdx.x * blockDim.x + threadIdx.x;
    if (i < n) deg[i] = rsqrtf(deg[i]);    // deg >= 1 always (self loops)
}

// ---------------- dense GEMM via V_WMMA_F32_16X16X4_F32 ----------------
// One wave32 per 16x16 output tile. K = DCH = 128 -> 32 chained WMMAs.
// Launched with exactly (Mtiles*Ntiles) waves so EXEC is all-ones (WMMA req).

__global__ void gcn_gemm_wmma(const float* __restrict__ A,   // [N, 128]
                              const float* __restrict__ W,   // [128, 128]
                              float* __restrict__ H) {       // [N, 128]
    int waveId = blockIdx.x * (blockDim.x >> 5) + (threadIdx.x >> 5);
    int lane   = threadIdx.x & 31;
    int mTile  = waveId >> 3;      // / (128/16)
    int nTile  = waveId & 7;
    int hi     = lane >> 4;        // lane half: selects K pair
    int l16    = lane & 15;

    int row = mTile * 16 + l16;    // A row held by this lane
    int col = nTile * 16 + l16;    // B/C column held by this lane

    const float* arow = A + (size_t)row * DCH + 2 * hi;

    v8f c = {};
#pragma unroll
    for (int k0 = 0; k0 < DCH; k0 += 4) {
        v2f a, b;
        a.x = arow[k0 + 0];                    // K = k0 + 2*hi
        a.y = arow[k0 + 1];                    // K = k0 + 2*hi + 1
        int kb = k0 + 2 * hi;
        b.x = W[(size_t)kb * DCH + col];       // B[K][col]
        b.y = W[(size_t)(kb + 1) * DCH + col];
        c = __builtin_amdgcn_wmma_f32_16x16x4_f32(
                /*neg_a=*/false, a, /*neg_b=*/false, b,
                /*c_mod=*/(short)0, c, /*reuse_a=*/false, /*reuse_b=*/false);
    }

    float* hcol = H + (size_t)(mTile * 16) * DCH + nTile * 16 + l16;
#pragma unroll
    for (int i = 0; i < 8; ++i)
        hcol[(size_t)(i + 8 * hi) * DCH] = c[i];   // M = i + 8*hi
}

// ---------------- aggregation ----------------

__global__ void gcn_zero(float* p, int n) {
    int i = blockIdx.x * blockDim.x + threadIdx.x;
    if (i < n) p[i] = 0.0f;
}

// one wave32 per edge; lane moves float4 (128 ch / 32 lanes = 4)
__global__ void gcn_scatter(const float* __restrict__ H,
                            const int* __restrict__ src,
                            const int* __restrict__ tgt,
                            const float* __restrict__ dis,
                            float* __restrict__ agg, int e) {
    int wave = blockIdx.x * (blockDim.x >> 5) + (threadIdx.x >> 5);
    int lane = threadIdx.x & 31;
    if (wave >= e) return;
    int s = src[wave];
    int t = tgt[wave];
    float w = dis[s] * dis[t];
    float4 v = ((const float4*)(H + (size_t)s * DCH))[lane];
    float* at = agg + (size_t)t * DCH + lane * 4;
    atomicAdd(at + 0, v.x * w);
    atomicAdd(at + 1, v.y * w);
    atomicAdd(at + 2, v.z * w);
    atomicAdd(at + 3, v.w * w);
}

// out = (agg + H*dis^2 (self loop) + bias), optional ReLU
__global__ void gcn_finalize(const float* __restrict__ agg,
                             const float* __restrict__ H,
                             const float* __restrict__ dis,
                             const float* __restrict__ bias,
                             float* __restrict__ out, int nd, int relu) {
    int i = blockIdx.x * blockDim.x + threadIdx.x;
    if (i >= nd) return;
    int node = i >> 7;          // / 128
    int d    = i & (DCH - 1);
    float di = dis[node];
    float v  = agg[i] + H[i] * di * di + bias[d];
    out[i] = relu ? fmaxf(v, 0.0f) : v;
}

// ---------------- driver ----------------

extern "C" void kernel_launch(void* const* d_in, const int* in_sizes, int n_in,
                              void* d_out, int out_size, void* d_ws, size_t ws_size,
                              hipStream_t stream) {
    const float* X  = (const float*)d_in[0];
    const int*   ei = (const int*)  d_in[1];   // [2, E]: row0 = src, row1 = tgt
    const float* W1 = (const float*)d_in[2];
    const float* b1 = (const float*)d_in[3];
    const float* W2 = (const float*)d_in[4];
    const float* b2 = (const float*)d_in[5];
    const float* W3 = (const float*)d_in[6];
    const float* b3 = (const float*)d_in[7];
    float* out = (float*)d_out;

    const int n  = in_sizes[0] / DCH;      // 50000
    const int e  = in_sizes[1] / 2;        // 800000
    const int nd = n * DCH;

    const int* src = ei;
    const int* tgt = ei + e;

    // workspace: dis[n] | H[n*128] | AGG[n*128]
    float* dis = (float*)d_ws;
    float* H   = dis + n;
    float* AGG = H + nd;

    // --- normalization ---
    gcn_deg_init <<<(n + 255) / 256, 256, 0, stream>>>(dis, n);
    gcn_deg_count<<<(e + 255) / 256, 256, 0, stream>>>(tgt, dis, e);
    gcn_rsqrt    <<<(n + 255) / 256, 256, 0, stream>>>(dis, n);

    const int mTiles     = n / 16;                 // 3125 (n divisible by 16)
    const int gemmBlocks = (mTiles * 8) / 8;       // 8 waves (tiles) per block
    const int edgeBlocks = (e + 7) / 8;            // 8 waves (edges) per block
    const int ndBlocks   = (nd + 255) / 256;

    const float* Ws[3] = {W1, W2, W3};
    const float* bs[3] = {b1, b2, b3};
    const float* cur   = X;

    for (int layer = 0; layer < 3; ++layer) {
        gcn_gemm_wmma<<<gemmBlocks, 256, 0, stream>>>(cur, Ws[layer], H);
        gcn_zero     <<<ndBlocks,   256, 0, stream>>>(AGG, nd);
        gcn_scatter  <<<edgeBlocks, 256, 0, stream>>>(H, src, tgt, dis, AGG, e);
        gcn_finalize <<<ndBlocks,   256, 0, stream>>>(AGG, H, dis, bs[layer], out,
                                                      nd, layer < 2 ? 1 : 0);
        cur = out;   // layer output ping-pongs through d_out (safe: GEMM of the
                     // next layer reads it before its finalize overwrites it)
    }
}